// Align_adaK_14027363188934
// MI455X (gfx1250) — compile-verified
//
#include <hip/hip_runtime.h>
#include <hip/hip_bf16.h>

typedef __attribute__((ext_vector_type(2))) float v2f;
typedef __attribute__((ext_vector_type(8))) float v8f;

// ---------------------------------------------------------------------------
// Shapes
//   B=16, T=32, C=128, CIN=2048, HW=7, NCLS=20, K = CIN*9 = 18432
// Workspace layout (floats):
//   tokpart [2][4][512][128]          @ 0        (524288)
//   tok     [2][512][128]             @ 524288   (131072)
//   f       [2][16][32][128]          @ 655360   (131072)
//   P       [16][32][32]              @ 786432   (16384)
//   Dall    [32][16][32][32]          @ 802816   (524288)
//   Dind    [32][16][32][32] (int)    @ 1327104  (524288)
//   sidx1/sidx2/smask [16][32] (int)  @ 1851392/1851904/1852416
//   partial [32]                      @ 1852928
// Total ~7.4 MB.
// ---------------------------------------------------------------------------

#define CI_CHUNK 64
#define KC (CI_CHUNK * 9)   // 576
#define SROW (KC + 4)       // 580: stride%64==4 -> conflict-free A-frag reads

// Fused 9-tap average pooling (mean of SAME-conv == weighted window sums) + WMMA GEMM.
__global__ __launch_bounds__(256)
void pool_gemm_kernel(const float* __restrict__ x1, const float* __restrict__ x2,
                      const float* __restrict__ w_bn, float* __restrict__ tokpart) {
  const int mtile = blockIdx.x;   // 0..31  (16 rows of bt each)
  const int kslice = blockIdx.y;  // 0..3   (512 input channels each)
  const int inp = blockIdx.z;     // 0..1
  const float* __restrict__ x = inp ? x2 : x1;
  const int bt0 = mtile * 16;
  const int tid = threadIdx.x;
  const int lane = tid & 31;
  const int wv = tid >> 5;        // wave -> N tile (8 waves x 16 cols = 128)
  const int mrow = lane & 15;
  const int half = lane >> 4;

  __shared__ float S[16 * SROW];

  v8f acc = {0.f, 0.f, 0.f, 0.f, 0.f, 0.f, 0.f, 0.f};

  const int pm = tid >> 4;          // pooled row 0..15
  const int pc0 = (tid & 15) * 4;   // 4 channels per thread per chunk

  for (int cc = 0; cc < 8; ++cc) {
    const int ci0 = kslice * 512 + cc * CI_CHUNK;
    __syncthreads();
    // ---- pooling phase: 16 x 64 images of 7x7 -> 9 shifted sums each ----
    for (int q = 0; q < 4; ++q) {
      const int cil = pc0 + q;
      const float* img = x + ((size_t)(bt0 + pm) * 2048 + (size_t)(ci0 + cil)) * 49;
      float F = 0.f, r0 = 0.f, r6 = 0.f, c0 = 0.f, c6 = 0.f;
      float v00 = 0.f, v06 = 0.f, v60 = 0.f, v66 = 0.f;
#pragma unroll
      for (int u = 0; u < 7; ++u) {
#pragma unroll
        for (int vv = 0; vv < 7; ++vv) {
          float t = img[u * 7 + vv];
          F += t;
          if (u == 0) r0 += t;
          if (u == 6) r6 += t;
          if (vv == 0) c0 += t;
          if (vv == 6) c6 += t;
          if (u == 0 && vv == 0) v00 = t;
          if (u == 0 && vv == 6) v06 = t;
          if (u == 6 && vv == 0) v60 = t;
          if (u == 6 && vv == 6) v66 = t;
        }
      }
      const float inv49 = 1.0f / 49.0f;
      float* dst = &S[pm * SROW + cil * 9];
#pragma unroll
      for (int kh = 0; kh < 3; ++kh) {
        float exR = (kh == 0) ? r6 : (kh == 2) ? r0 : 0.f;
#pragma unroll
        for (int kw = 0; kw < 3; ++kw) {
          float exC = (kw == 0) ? c6 : (kw == 2) ? c0 : 0.f;
          float corner = 0.f;
          if (kh == 0 && kw == 0) corner = v66;
          if (kh == 0 && kw == 2) corner = v60;
          if (kh == 2 && kw == 0) corner = v06;
          if (kh == 2 && kw == 2) corner = v00;
          dst[kh * 3 + kw] = (F - exR - exC + corner) * inv49;
        }
      }
    }
    __syncthreads();
    // ---- WMMA phase: D(16x16) += A(16x4,f32,LDS) x B(4x16,f32,w_bn^T) ----
    const int n = wv * 16 + mrow;                       // B col (lane%16)
    const size_t wrowbase = (size_t)n * 18432 + (size_t)ci0 * 9;
    for (int kl = 0; kl < KC; kl += 4) {
      // A element (m = lane%16, k = 2*(lane/16)+v)
      v2f a = *reinterpret_cast<const v2f*>(&S[mrow * SROW + kl + 2 * half]);
      // B element (k = 2*v + lane/16, n = lane%16)
      v2f bf;
      bf.x = w_bn[wrowbase + kl + half];
      bf.y = w_bn[wrowbase + kl + half + 2];
      acc = __builtin_amdgcn_wmma_f32_16x16x4_f32(
          false, a, false, bf, (short)0, acc, false, false);
    }
  }
  // D element (m = v + 8*(lane/16), n = lane%16)
  float* outp = tokpart + ((size_t)inp * 262144 + (size_t)kslice * 65536);
#pragma unroll
  for (int v = 0; v < 8; ++v) {
    int m = v + 8 * half;
    outp[(size_t)(bt0 + m) * 128 + wv * 16 + mrow] = acc[v];
  }
}

__global__ void tok_reduce_kernel(const float* __restrict__ tokpart,
                                  const float* __restrict__ b_bn,
                                  float* __restrict__ tok) {
  int idx = blockIdx.x * 256 + threadIdx.x;
  if (idx >= 131072) return;
  int inp = idx >> 16;
  int r = idx & 65535;
  int c = r & 127;
  const float* p = tokpart + (size_t)inp * 262144;
  tok[idx] = b_bn[c] + p[r] + p[65536 + r] + p[131072 + r] + p[196608 + r];
}

// conv1d(k=3,SAME)+ReLU then conv1d(k=1) over T, per batch, per input.
__global__ __launch_bounds__(256)
void conv1d_kernel(const float* __restrict__ tok, const float* __restrict__ w_s1,
                   const float* __restrict__ b_s1, const float* __restrict__ w_s2,
                   const float* __restrict__ b_s2, float* __restrict__ f) {
  const int b = blockIdx.x, inp = blockIdx.y;
  const float* tk = tok + (size_t)inp * 65536 + (size_t)b * 4096;
  float* fo = f + (size_t)inp * 65536 + (size_t)b * 4096;
  __shared__ float tokS[4096];
  __shared__ float h1S[4096];
  const int tid = threadIdx.x;
  for (int p = tid; p < 4096; p += 256) tokS[p] = tk[p];
  __syncthreads();
  for (int p = tid; p < 4096; p += 256) {
    int t = p >> 7, c = p & 127;
    float acc = b_s1[c];
#pragma unroll
    for (int dt = 0; dt < 3; ++dt) {
      int tt = t + dt - 1;
      if (tt < 0 || tt > 31) continue;
      const float* wr = w_s1 + (size_t)c * 384 + dt;
      const float* tr = tokS + tt * 128;
      for (int ci = 0; ci < 128; ++ci) acc += wr[ci * 3] * tr[ci];
    }
    h1S[p] = acc > 0.f ? acc : 0.f;
  }
  __syncthreads();
  for (int p = tid; p < 4096; p += 256) {
    int t = p >> 7, c = p & 127;
    float acc = b_s2[c];
    const float* wr = w_s2 + (size_t)c * 128;
    const float* hr = h1S + t * 128;
    for (int ci = 0; ci < 128; ++ci) acc += wr[ci] * hr[ci];
    fo[p] = acc;
  }
}

// logits -> softmax rows -> 2D cumsum P -> D0 = P/((i+1)(j+1))
__global__ __launch_bounds__(256)
void predP_kernel(const float* __restrict__ f1g, const float* __restrict__ f2g,
                  float* __restrict__ P, float* __restrict__ D0) {
  const int b = blockIdx.x;
  __shared__ float F1[4096], F2[4096];
  __shared__ float M[32 * 33];
  const int tid = threadIdx.x;
  for (int p = tid; p < 4096; p += 256) {
    F1[p] = f1g[(size_t)b * 4096 + p];
    F2[p] = f2g[(size_t)b * 4096 + p];
  }
  __syncthreads();
  for (int p = tid; p < 1024; p += 256) {
    int i = p >> 5, j = p & 31;
    float d = 0.f;
    for (int c = 0; c < 128; ++c) d += F1[i * 128 + c] * F2[j * 128 + c];
    M[i * 33 + j] = d * 0.08838834764831845f;  // 1/sqrt(128)
  }
  __syncthreads();
  {
    int w = tid >> 5, lane = tid & 31;
    for (int i = w; i < 32; i += 8) {
      float v = M[i * 33 + lane];
      float mx = v;
      for (int off = 16; off; off >>= 1) mx = fmaxf(mx, __shfl_xor(mx, off, 32));
      float e = __expf(v - mx);
      float s = e;
      for (int off = 16; off; off >>= 1) s += __shfl_xor(s, off, 32);
      M[i * 33 + lane] = e / s;
    }
  }
  __syncthreads();
  if (tid < 32) { float r = 0.f; for (int j = 0; j < 32; ++j) { r += M[tid * 33 + j]; M[tid * 33 + j] = r; } }
  __syncthreads();
  if (tid < 32) { float r = 0.f; for (int i = 0; i < 32; ++i) { r += M[i * 33 + tid]; M[i * 33 + tid] = r; } }
  __syncthreads();
  for (int p = tid; p < 1024; p += 256) {
    int i = p >> 5, j = p & 31;
    float pv = M[i * 33 + j];
    P[(size_t)b * 1024 + p] = pv;
    D0[(size_t)b * 1024 + p] = pv / (float)((i + 1) * (j + 1));
  }
}

// One DP step: Dk[b,i,j] = max_{a<i,bb<j} Dprev[a,bb] + (Pij-Pib-Paj+Pab)/((i-a)(j-bb))
__global__ __launch_bounds__(256)
void dp_step_kernel(const float* __restrict__ P, float* __restrict__ Dall,
                    int* __restrict__ Dind, int k) {
  const int i = blockIdx.x, b = blockIdx.y;
  __shared__ float Dp[1024], Ps[1024], invT[32];
  __shared__ float rv[256];
  __shared__ int ri[256];
  const int tid = threadIdx.x;
  const float* Dprev = Dall + (size_t)(k - 1) * 16384 + (size_t)b * 1024;
  for (int p = tid; p < 1024; p += 256) { Dp[p] = Dprev[p]; Ps[p] = P[(size_t)b * 1024 + p]; }
  if (tid < 32) invT[tid] = tid ? 1.0f / (float)tid : 0.f;
  __syncthreads();
  const int j = tid >> 3, sub = tid & 7;
  const float Pij = Ps[i * 32 + j];
  float best = -__builtin_huge_valf();
  int bidx = 0;
  for (int a = sub; a < i; a += 8) {
    const float Paj = Ps[a * 32 + j];
    const float ia = invT[i - a];
    const int base = a * 32;
    for (int bb = 0; bb < j; ++bb) {
      float blk = (Pij - Ps[i * 32 + bb] - Paj + Ps[base + bb]) * ia * invT[j - bb];
      float val = Dp[base + bb] + blk;
      if (val > best) { best = val; bidx = base + bb; }
    }
  }
  rv[tid] = best; ri[tid] = bidx;
  __syncthreads();
  if (sub == 0) {
    for (int s = 1; s < 8; ++s) {
      float v = rv[tid + s]; int id = ri[tid + s];
      if (v > best || (v == best && id < bidx)) { best = v; bidx = id; }
    }
    Dall[(size_t)k * 16384 + (size_t)b * 1024 + i * 32 + j] = best;
    Dind[(size_t)k * 16384 + (size_t)b * 1024 + i * 32 + j] = bidx;
  }
}

__global__ void backtrace_kernel(const float* __restrict__ Dall, const int* __restrict__ Dind,
                                 int* __restrict__ sidx1, int* __restrict__ sidx2,
                                 int* __restrict__ smask, float* __restrict__ partial,
                                 float* __restrict__ out) {
  __shared__ float maxs[16];
  const int tid = threadIdx.x;
  if (tid < 16) {
    const int b = tid;
    float best = -__builtin_huge_valf();
    int k0 = 0;
    for (int k = 0; k < 32; ++k) {
      float v = Dall[(size_t)k * 16384 + (size_t)b * 1024 + 1023];
      if (v > best) { best = v; k0 = k; }
    }
    maxs[b] = best;
    int i = 31, j = 31, k = k0;
    sidx1[b * 32] = 1; sidx2[b * 32] = 1; smask[b * 32] = 1;
    for (int s = 0; s < 31; ++s) {
      int act = (k > 0);
      int ind = act ? Dind[(size_t)k * 16384 + (size_t)b * 1024 + i * 32 + j] : 0;
      int a = ind >> 5, bv = ind & 31;
      int i1 = a + 1; if (i1 > 31) i1 = 31;
      int j1 = bv + 1; if (j1 > 31) j1 = 31;
      sidx1[b * 32 + s + 1] = act ? i1 : 0;
      sidx2[b * 32 + s + 1] = act ? j1 : 0;
      smask[b * 32 + s + 1] = act;
      if (act) { i = a; j = bv; k = k - 1; }
    }
  }
  if (tid < 32) partial[tid] = 0.f;
  __syncthreads();
  if (tid == 0) {
    float s = 0.f;
    for (int b = 0; b < 16; ++b) s += maxs[b];
    out[641] = -s / 16.0f;  // loss_step
  }
}

__global__ __launch_bounds__(256)
void f2s_kernel(const float* __restrict__ f1g, const float* __restrict__ f2g,
                const int* __restrict__ sidx1, const int* __restrict__ sidx2,
                const int* __restrict__ smask, float* __restrict__ partial) {
  const int b = blockIdx.x, dir = blockIdx.y;
  const float* frames = dir ? f2g : f1g;
  const float* src = dir ? f1g : f2g;
  const int* sidx = dir ? sidx1 : sidx2;
  __shared__ float St[4096];
  __shared__ float frow[128];
  __shared__ float part[256];
  __shared__ float d2s[32];
  __shared__ int mk[32];
  __shared__ float tsum;
  const int tid = threadIdx.x;
  for (int p = tid; p < 4096; p += 256) {
    int s = p >> 7, c = p & 127;
    St[p] = src[((size_t)b * 32 + sidx[b * 32 + s]) * 128 + c];
  }
  if (tid < 32) mk[tid] = smask[b * 32 + tid];
  if (tid == 0) tsum = 0.f;
  __syncthreads();
  for (int t = 0; t < 32; ++t) {
    if (tid < 128) frow[tid] = frames[((size_t)b * 32 + t) * 128 + tid];
    __syncthreads();
    int s = tid >> 3, ch = (tid & 7) * 16;
    float pp = 0.f;
    for (int c = ch; c < ch + 16; ++c) { float d = frow[c] - St[s * 128 + c]; pp += d * d; }
    part[tid] = pp;
    __syncthreads();
    if (tid < 32) {
      float q = 0.f;
      for (int r = 0; r < 8; ++r) q += part[tid * 8 + r];
      d2s[tid] = mk[tid] ? q : __builtin_huge_valf();
    }
    __syncthreads();
    if (tid == 0) {
      float mn = d2s[0];
      for (int s3 = 1; s3 < 32; ++s3) mn = fminf(mn, d2s[s3]);
      tsum += mn;
    }
    __syncthreads();
  }
  if (tid == 0) partial[dir * 16 + b] = tsum;
}

__global__ void finalize_kernel(const float* __restrict__ partial, float* __restrict__ out) {
  if (threadIdx.x == 0) {
    float s = 0.f;
    for (int p = 0; p < 32; ++p) s += partial[p];
    out[640] = s / 512.0f;  // f2s = mean1 + mean2, each over B*T=512
  }
}

__global__ __launch_bounds__(128)
void head_kernel(const float* __restrict__ f, const float* __restrict__ w_g,
                 const float* __restrict__ b_g, const float* __restrict__ w_c,
                 const float* __restrict__ b_c, float* __restrict__ out) {
  const int b = blockIdx.x, inp = blockIdx.y;
  const float* fr = f + (size_t)inp * 65536 + (size_t)b * 4096;
  __shared__ float g[128];
  const int c = threadIdx.x;
  float acc = b_g[c];
  for (int k = 0; k < 4096; ++k) acc += fr[k] * w_g[(size_t)k * 128 + c];
  g[c] = acc;
  __syncthreads();
  if (c < 20) {
    float p = b_c[c];
    for (int cc = 0; cc < 128; ++cc) p += g[cc] * w_c[cc * 20 + c];
    out[(size_t)inp * 320 + b * 20 + c] = p;
  }
}

extern "C" void kernel_launch(void* const* d_in, const int* in_sizes, int n_in,
                              void* d_out, int out_size, void* d_ws, size_t ws_size,
                              hipStream_t stream) {
  const float* x1 = (const float*)d_in[0];
  const float* x2 = (const float*)d_in[1];
  const float* w_bn = (const float*)d_in[2];
  const float* b_bn = (const float*)d_in[3];
  const float* w_s1 = (const float*)d_in[4];
  const float* b_s1 = (const float*)d_in[5];
  const float* w_s2 = (const float*)d_in[6];
  const float* b_s2 = (const float*)d_in[7];
  const float* w_g = (const float*)d_in[8];
  const float* b_g = (const float*)d_in[9];
  const float* w_c = (const float*)d_in[10];
  const float* b_c = (const float*)d_in[11];
  float* out = (float*)d_out;
  float* ws = (float*)d_ws;

  float* tokpart = ws;                    // [2][4][512][128]
  float* tok     = ws + 524288;           // [2][512][128]
  float* f       = ws + 655360;           // [2][16][32][128]
  float* P       = ws + 786432;           // [16][32][32]
  float* Dall    = ws + 802816;           // [32][16][32][32]
  int*   Dind    = (int*)(ws + 1327104);  // [32][16][32][32]
  int*   sidx1   = (int*)(ws + 1851392);
  int*   sidx2   = (int*)(ws + 1851904);
  int*   smask   = (int*)(ws + 1852416);
  float* partial = ws + 1852928;          // [32]

  pool_gemm_kernel<<<dim3(32, 4, 2), 256, 0, stream>>>(x1, x2, w_bn, tokpart);
  tok_reduce_kernel<<<512, 256, 0, stream>>>(tokpart, b_bn, tok);
  conv1d_kernel<<<dim3(16, 2), 256, 0, stream>>>(tok, w_s1, b_s1, w_s2, b_s2, f);
  predP_kernel<<<16, 256, 0, stream>>>(f, f + 65536, P, Dall);
  for (int k = 1; k < 32; ++k)
    dp_step_kernel<<<dim3(32, 16), 256, 0, stream>>>(P, Dall, Dind, k);
  backtrace_kernel<<<1, 64, 0, stream>>>(Dall, Dind, sidx1, sidx2, smask, partial, out);
  f2s_kernel<<<dim3(16, 2), 256, 0, stream>>>(f, f + 65536, sidx1, sidx2, smask, partial);
  finalize_kernel<<<1, 32, 0, stream>>>(partial, out);
  head_kernel<<<dim3(16, 2), 128, 0, stream>>>(f, w_g, b_g, w_c, b_c, out);
}